// L3AGNNConv_84859963834423
// MI455X (gfx1250) — compile-verified
//
#include <hip/hip_runtime.h>
#include <math.h>

#define NN   50000
#define EE   800000
#define ETOT (EE + NN)
#define D1   200
#define D2   100
#define D3   2
#define K1   128
#define EPSF 1e-12f
#define KEEP 0.9f

typedef __attribute__((ext_vector_type(2))) float v2f;
typedef __attribute__((ext_vector_type(8))) float v8f;

// ---------------- W pre-pack ----------------
// WP[((k>>1)*Dp + col)*2 + (k&1)] = (col < D) ? W[k*D + col] : 0
// so a wave's B fragment for each WMMA k-step is one aligned b64 load.
template <int K, int D>
__global__ void agnn_pack_w(const float* __restrict__ W, float* __restrict__ WP) {
    constexpr int Dp = (D + 15) & ~15;
    int i = blockIdx.x * blockDim.x + threadIdx.x;
    if (i >= K * Dp) return;
    int k = i / Dp, col = i - k * Dp;
    float v = (col < D) ? W[k * D + col] : 0.0f;
    WP[(((k >> 1) * Dp + col) << 1) + (k & 1)] = v;
}

// ---------------- GEMM via fp32 WMMA (16x16x4) ----------------
// One wave computes one 16x16 tile of H = X(N x K) * W(K x D), K-loop step 4.
// A layout (16x4 f32): lane l -> M = l&15 ; VGPR0/1 = K(2*(l>>4)), K(2*(l>>4)+1)
// B layout (4x16 f32): lane l -> N = l&15 ; same K split across lane halves
// C/D layout: VGPR i : lanes 0-15 -> M=i, lanes 16-31 -> M=i+8 ; N = l&15
template <int K, int D>
__global__ void agnn_gemm_wmma_f32(const float* __restrict__ X,
                                   const float* __restrict__ WP,
                                   float* __restrict__ H) {
    constexpr int Dp = (D + 15) & ~15;
    constexpr int tiles_n = Dp >> 4;
    constexpr int tiles_m = NN >> 4;
    int wave = threadIdx.x >> 5;
    int lane = threadIdx.x & 31;
    int tile = blockIdx.x * (blockDim.x >> 5) + wave;
    if (tile >= tiles_m * tiles_n) return;
    int tm = tile / tiles_n, tn = tile - tm * tiles_n;
    int m0 = tm << 4, n0 = tn << 4;
    int ml = lane & 15, kg = lane >> 4;                 // kg in {0,1}
    int col = n0 + ml;                                  // col < Dp always

    const float* ap = X + (size_t)(m0 + ml) * K + 2 * kg;   // + 4 floats / iter
    const float* bp = WP + (((size_t)kg * Dp + col) << 1);  // + 4*Dp floats / iter

    v8f c = {};
    #pragma unroll 8
    for (int k = 0; k < K; k += 4) {
        v2f a = *(const v2f*)(ap + k);
        v2f b = *(const v2f*)(bp + (size_t)k * Dp);
        c = __builtin_amdgcn_wmma_f32_16x16x4_f32(false, a, false, b,
                                                  (short)0, c, false, false);
    }
    if (col < D) {
        int rbase = m0 + 8 * kg;
        #pragma unroll
        for (int i = 0; i < 8; ++i)
            H[(size_t)(rbase + i) * D + col] = c[i];
    }
}

// ---------------- row inverse-norm (wave per node) ----------------
template <int D>
__global__ void agnn_rownorm(const float* __restrict__ H, float* __restrict__ inv) {
    int node = (int)((blockIdx.x * (size_t)blockDim.x + threadIdx.x) >> 5);
    int lane = threadIdx.x & 31;
    if (node >= NN) return;
    const float* row = H + (size_t)node * D;
    float s = 0.0f;
    #pragma unroll
    for (int d = lane; d < D; d += 32) { float v = row[d]; s += v * v; }
    #pragma unroll
    for (int off = 16; off; off >>= 1) s += __shfl_xor(s, off, 32);
    if (lane == 0) inv[node] = 1.0f / fmaxf(sqrtf(s), EPSF);
}

// ---------------- fills ----------------
__global__ void fill_i32(int* __restrict__ p, int v, int n) {
    int i = blockIdx.x * blockDim.x + threadIdx.x;
    if (i < n) p[i] = v;
}
__global__ void fill_f32(float* __restrict__ p, float v, int n) {
    int i = blockIdx.x * blockDim.x + threadIdx.x;
    if (i < n) p[i] = v;
}

// order-preserving float <-> int map for atomicMax
__device__ __forceinline__ int fmap(float f) {
    int i = __float_as_int(f);
    return (i < 0) ? (i ^ 0x7fffffff) : i;
}
__device__ __forceinline__ float funmap(int i) {
    return __int_as_float((i < 0) ? (i ^ 0x7fffffff) : i);
}

// ---------------- edge logits + segment max (wave per edge) ----------------
template <int D>
__global__ void agnn_edge_logit(const float* __restrict__ H, const float* __restrict__ inv,
                                const int* __restrict__ ei, const float* __restrict__ beta,
                                float* __restrict__ elog, int* __restrict__ amax) {
    int e = (int)((blockIdx.x * (size_t)blockDim.x + threadIdx.x) >> 5);
    int lane = threadIdx.x & 31;
    if (e >= ETOT) return;
    int s, d;
    if (e < EE) { s = ei[e]; d = ei[EE + e]; } else { s = d = e - EE; }
    const float* hs = H + (size_t)s * D;
    const float* hd = H + (size_t)d * D;
    float acc = 0.0f;
    #pragma unroll
    for (int i = lane; i < D; i += 32) acc += hs[i] * hd[i];
    #pragma unroll
    for (int off = 16; off; off >>= 1) acc += __shfl_xor(acc, off, 32);
    if (lane == 0) {
        float a = beta[0] * acc * inv[s] * inv[d];
        elog[e] = a;
        atomicMax(&amax[d], fmap(a));
    }
}

// ---------------- exp + segment sum (thread per edge) ----------------
__global__ void agnn_edge_expsum(const int* __restrict__ ei, const int* __restrict__ amax,
                                 float* __restrict__ elog, float* __restrict__ denom) {
    int e = blockIdx.x * blockDim.x + threadIdx.x;
    if (e >= ETOT) return;
    int d = (e < EE) ? ei[EE + e] : (e - EE);
    float v = expf(elog[e] - funmap(amax[d]));
    elog[e] = v;
    atomicAdd(&denom[d], v);
}

// ---------------- weighted scatter-sum (wave per edge) ----------------
template <int D>
__global__ void agnn_edge_aggregate(const float* __restrict__ H, const int* __restrict__ ei,
                                    const float* __restrict__ elog,
                                    const float* __restrict__ denom,
                                    float* __restrict__ out) {
    int e = (int)((blockIdx.x * (size_t)blockDim.x + threadIdx.x) >> 5);
    int lane = threadIdx.x & 31;
    if (e >= ETOT) return;
    int s, d;
    if (e < EE) { s = ei[e]; d = ei[EE + e]; } else { s = d = e - EE; }
    float alpha = elog[e] / fmaxf(denom[d], EPSF);
    const float* hs = H + (size_t)s * D;
    float* od = out + (size_t)d * D;
    #pragma unroll
    for (int i = lane; i < D; i += 32)
        atomicAdd(&od[i], hs[i] * alpha);
}

// ---------------- relu (+ deterministic hash dropout) ----------------
__device__ __forceinline__ unsigned hash_u32(unsigned x) {
    x ^= x >> 16; x *= 0x7feb352dU;
    x ^= x >> 15; x *= 0x846ca68bU;
    x ^= x >> 16; return x;
}
__global__ void agnn_relu_dropout(float* __restrict__ p, int n, unsigned seed, int do_drop) {
    int i = blockIdx.x * blockDim.x + threadIdx.x;
    if (i >= n) return;
    float v = fmaxf(p[i], 0.0f);
    if (do_drop) {
        unsigned h = hash_u32((unsigned)i * 0x9E3779B9u + seed);
        const unsigned thresh = (unsigned)(KEEP * 4294967296.0);
        v = (h < thresh) ? v * (1.0f / KEEP) : 0.0f;
    }
    p[i] = v;
}

// ---------------- one AGNN layer ----------------
template <int K, int D>
static void run_layer(const float* X, const float* W, const float* beta,
                      float* WP, float* H, float* OUT, float* inv, int* amax,
                      float* denom, float* elog, const int* ei,
                      int do_drop, unsigned seed, hipStream_t stream) {
    constexpr int Dp = (D + 15) & ~15;
    agnn_pack_w<K, D><<<(K * Dp + 255) / 256, 256, 0, stream>>>(W, WP);
    constexpr int tiles = (NN / 16) * (Dp / 16);
    agnn_gemm_wmma_f32<K, D><<<(tiles + 7) / 8, 256, 0, stream>>>(X, WP, H);
    agnn_rownorm<D><<<(NN * 32 + 255) / 256, 256, 0, stream>>>(H, inv);
    fill_i32<<<(NN + 255) / 256, 256, 0, stream>>>(amax, (int)0x80000000, NN);
    fill_f32<<<(NN + 255) / 256, 256, 0, stream>>>(denom, 0.0f, NN);
    constexpr int outn = NN * D;
    fill_f32<<<(outn + 255) / 256, 256, 0, stream>>>(OUT, 0.0f, outn);
    agnn_edge_logit<D><<<(ETOT + 7) / 8, 256, 0, stream>>>(H, inv, ei, beta, elog, amax);
    agnn_edge_expsum<<<(ETOT + 255) / 256, 256, 0, stream>>>(ei, amax, elog, denom);
    agnn_edge_aggregate<D><<<(ETOT + 7) / 8, 256, 0, stream>>>(H, ei, elog, denom, OUT);
    agnn_relu_dropout<<<(outn + 255) / 256, 256, 0, stream>>>(OUT, outn, seed, do_drop);
}

extern "C" void kernel_launch(void* const* d_in, const int* in_sizes, int n_in,
                              void* d_out, int out_size, void* d_ws, size_t ws_size,
                              hipStream_t stream) {
    const float* x  = (const float*)d_in[0];
    const int*   ei = (const int*)d_in[1];   // [2, E] : row0 = src, row1 = dst
    const float* W1 = (const float*)d_in[2];
    const float* W2 = (const float*)d_in[3];
    const float* W3 = (const float*)d_in[4];
    const float* b1 = (const float*)d_in[5];
    const float* b2 = (const float*)d_in[6];
    const float* b3 = (const float*)d_in[7];

    // workspace (floats):
    // hbuf[N*200] | obuf[N*200] | inv[N] | denom[N] | amax[N] | elog[E+N] | WP[max K*Dp]
    const size_t WPMAX = (size_t)D1 * 208;   // >= 128*208, 200*112, 100*16
    size_t need = ((size_t)NN * D1 * 2 + (size_t)NN * 3 + (size_t)ETOT + WPMAX) * sizeof(float);
    if (ws_size < need) return;
    float* ws    = (float*)d_ws;
    float* hbuf  = ws;
    float* obuf  = hbuf + (size_t)NN * D1;
    float* inv   = obuf + (size_t)NN * D1;
    float* denom = inv + NN;
    int*   amax  = (int*)(denom + NN);
    float* elog  = (float*)(amax + NN);
    float* WP    = elog + ETOT;

    // layer 1: 128 -> 200, relu + dropout
    run_layer<K1, D1>(x,    W1, b1, WP, hbuf, obuf, inv, amax, denom, elog, ei, 1, 0x1234567u, stream);
    // layer 2: 200 -> 100, relu + dropout  (obuf reused in-place: GEMM consumes it first)
    run_layer<D1, D2>(obuf, W2, b2, WP, hbuf, obuf, inv, amax, denom, elog, ei, 1, 0x89ABCDEu, stream);
    // layer 3: 100 -> 2, relu only, straight into d_out
    run_layer<D2, D3>(obuf, W3, b3, WP, hbuf, (float*)d_out, inv, amax, denom, elog, ei, 0, 0u, stream);
    (void)in_sizes; (void)n_in; (void)out_size;
}